// Pam_2748779070022
// MI455X (gfx1250) — compile-verified
//
#include <hip/hip_runtime.h>
#include <hip/hip_bf16.h>

typedef float v2f __attribute__((ext_vector_type(2)));
typedef float v8f __attribute__((ext_vector_type(8)));
typedef int   vi4 __attribute__((vector_size(16)));    // matches builtin param type

#define AS1 __attribute__((address_space(1)))
#define AS3 __attribute__((address_space(3)))

#if __has_builtin(__builtin_amdgcn_global_load_async_to_lds_b128) && \
    __has_builtin(__builtin_amdgcn_global_load_async_to_lds_b32)
#define USE_ASYNC_LDS 1
#else
#define USE_ASYNC_LDS 0
#endif

#if USE_ASYNC_LDS
__device__ __forceinline__ void async_copy_b128(const float* g, float* l) {
    __builtin_amdgcn_global_load_async_to_lds_b128((AS1 vi4*)g, (AS3 vi4*)l, 0, 0);
}
__device__ __forceinline__ void async_copy_b32(const float* g, float* l) {
    __builtin_amdgcn_global_load_async_to_lds_b32((AS1 int*)g, (AS3 int*)l, 0, 0);
}
#endif

__device__ __forceinline__ void wait_async0() {
#if __has_builtin(__builtin_amdgcn_s_wait_asynccnt)
    __builtin_amdgcn_s_wait_asynccnt(0);
#else
    asm volatile("s_wait_asynccnt 0x0" ::: "memory");
#endif
}

// WG computes a 64(M) x 128(N) output tile, K chunked by 32, staged in LDS.
// 8 wave32s in a 2(M) x 4(N) arrangement; each wave owns four 16x16
// V_WMMA_F32_16X16X4_F32 accumulator tiles (32 WMMAs per k-chunk).
//
// Both LDS tiles are stored [row][k] (k contiguous, +4 pad) so every WMMA
// fragment read is a contiguous k-pair (ds_load b64), and the dominant
// out-GEMM stages both operands as direct 16B async copies.
//
// Operand views:
//   A element (m,k): a_k_minor ? A[m*a_major + k] : A[k*a_major + m]
//   B element (k,n): b_k_minor ? B[n*b_major + k] : B[k*b_major + n]

#define TM 64
#define TN 128
#define TK 32
#define PAD 4

__global__ void wmma_gemm_tiled(
    const float* __restrict__ A, long long a_major, long long a_batch, int a_k_minor,
    const float* __restrict__ B, long long b_major, long long b_batch, int b_k_minor,
    float* __restrict__ Y, long long y_batch,
    const float* __restrict__ bias,
    const float* __restrict__ resid, long long r_batch,
    int O, int K, int N, int NB)
{
    __shared__ float As[TM][TK + PAD];   // [m][k]
    __shared__ float Bs[TN][TK + PAD];   // [n][k]

    const int tid  = (int)threadIdx.x;           // 0..255
    const int wave = tid >> 5;                   // 0..7
    const int lane = tid & 31;
    const int h    = lane >> 4;                  // half-wave
    const int l    = lane & 15;
    const int koff = h * 2;                      // K offset within a 4-chunk

    const int bp   = (O / TM) * (N / TN);
    const int b    = (int)blockIdx.x / bp;
    int rem        = (int)blockIdx.x - b * bp;
    const int ot   = rem / (N / TN);
    const int nt   = rem - ot * (N / TN);
    const int m_blk = ot * TM;
    const int n_blk = nt * TN;

    const float* Ab = A + (long long)b * a_batch;
    const float* Bb = B + (long long)b * b_batch;

    const int wm = wave & 1;                     // 32-row sub-block
    const int wn = wave >> 1;                    // 32-col sub-block

    v8f acc00 = {}, acc01 = {}, acc10 = {}, acc11 = {};

    for (int k0 = 0; k0 < K; k0 += TK) {
        __syncthreads();                         // protect LDS from previous chunk

        // ---- stage A tile: As[m][k] ----
        if (a_k_minor) {
            // memory rows = m, contiguous k : 64 rows x 8 x 16B, direct copy
            for (int i = tid; i < TM * (TK / 4); i += 256) {
                const int row = i >> 3;
                const int c4  = (i & 7) * 4;
                const float* g = Ab + (long long)(m_blk + row) * a_major + k0 + c4;
#if USE_ASYNC_LDS
                async_copy_b128(g, &As[row][c4]);
#else
                const float4 v = *(const float4*)g;
                As[row][c4 + 0] = v.x; As[row][c4 + 1] = v.y;
                As[row][c4 + 2] = v.z; As[row][c4 + 3] = v.w;
#endif
            }
        } else {
            // memory rows = k, contiguous m : transpose into LDS
#if USE_ASYNC_LDS
            for (int i = tid; i < TK * TM; i += 256) {      // lanes walk contiguous m
                const int m    = i & (TM - 1);
                const int krow = i >> 6;
                async_copy_b32(Ab + (long long)(k0 + krow) * a_major + m_blk + m,
                               &As[m][krow]);
            }
#else
            for (int i = tid; i < TK * (TM / 4); i += 256) {
                const int krow = i >> 4;
                const int m4   = (i & 15) * 4;
                const float4 v = *(const float4*)(Ab + (long long)(k0 + krow) * a_major + m_blk + m4);
                As[m4 + 0][krow] = v.x; As[m4 + 1][krow] = v.y;
                As[m4 + 2][krow] = v.z; As[m4 + 3][krow] = v.w;
            }
#endif
        }

        // ---- stage B tile: Bs[n][k] ----
        if (b_k_minor) {
            // memory rows = n, contiguous k : 128 rows x 8 x 16B, direct copy
            for (int i = tid; i < TN * (TK / 4); i += 256) {
                const int nrow = i >> 3;
                const int k4   = (i & 7) * 4;
                const float* g = Bb + (long long)(n_blk + nrow) * b_major + k0 + k4;
#if USE_ASYNC_LDS
                async_copy_b128(g, &Bs[nrow][k4]);
#else
                const float4 v = *(const float4*)g;
                Bs[nrow][k4 + 0] = v.x; Bs[nrow][k4 + 1] = v.y;
                Bs[nrow][k4 + 2] = v.z; Bs[nrow][k4 + 3] = v.w;
#endif
            }
        } else {
            // memory rows = k, contiguous n : transpose into LDS
#if USE_ASYNC_LDS
            for (int i = tid; i < TK * TN; i += 256) {      // lanes walk contiguous n
                const int n    = i & (TN - 1);
                const int krow = i >> 7;
                async_copy_b32(Bb + (long long)(k0 + krow) * b_major + n_blk + n,
                               &Bs[n][krow]);
            }
#else
            for (int i = tid; i < TK * (TN / 4); i += 256) {
                const int krow = i >> 5;
                const int n4   = (i & 31) * 4;
                const float4 v = *(const float4*)(Bb + (long long)(k0 + krow) * b_major + n_blk + n4);
                Bs[n4 + 0][krow] = v.x; Bs[n4 + 1][krow] = v.y;
                Bs[n4 + 2][krow] = v.z; Bs[n4 + 3][krow] = v.w;
            }
#endif
        }

#if USE_ASYNC_LDS
        wait_async0();
#endif
        __syncthreads();

        // ---- compute: 8 k-steps x 4 WMMAs, all fragments contiguous k-pairs ----
        const int ma0 = wm * 32 + l;
        const int ma1 = wm * 32 + 16 + l;
        const int nb0 = wn * 32 + l;
        const int nb1 = wn * 32 + 16 + l;
#pragma unroll
        for (int ks = 0; ks < TK; ks += 4) {
            const int kk = ks + koff;
            v2f a0, a1, b0, b1;
            a0.x = As[ma0][kk]; a0.y = As[ma0][kk + 1];
            a1.x = As[ma1][kk]; a1.y = As[ma1][kk + 1];
            b0.x = Bs[nb0][kk]; b0.y = Bs[nb0][kk + 1];
            b1.x = Bs[nb1][kk]; b1.y = Bs[nb1][kk + 1];
            acc00 = __builtin_amdgcn_wmma_f32_16x16x4_f32(false, a0, false, b0, (short)0, acc00, false, false);
            acc01 = __builtin_amdgcn_wmma_f32_16x16x4_f32(false, a0, false, b1, (short)0, acc01, false, false);
            acc10 = __builtin_amdgcn_wmma_f32_16x16x4_f32(false, a1, false, b0, (short)0, acc10, false, false);
            acc11 = __builtin_amdgcn_wmma_f32_16x16x4_f32(false, a1, false, b1, (short)0, acc11, false, false);
        }
    }

    // ---- epilogue: C/D layout -> row (r + h*8), col l per 16x16 tile ----
    float* Yb = Y + (long long)b * y_batch;
    const float* Rb = resid ? (resid + (long long)b * r_batch) : nullptr;

#pragma unroll
    for (int tm = 0; tm < 2; ++tm) {
#pragma unroll
        for (int tn = 0; tn < 2; ++tn) {
            const v8f* ap = (tm == 0) ? ((tn == 0) ? &acc00 : &acc01)
                                      : ((tn == 0) ? &acc10 : &acc11);
#pragma unroll
            for (int r = 0; r < 8; ++r) {
                const int row = m_blk + wm * 32 + tm * 16 + r + h * 8;
                const int col = n_blk + wn * 32 + tn * 16 + l;
                float val = (*ap)[r];
                if (bias)  val += bias[row];
                const long long idx = (long long)row * N + col;
                if (Rb) val += Rb[idx];
                Yb[idx] = val;
            }
        }
    }
}

// ---------------------------------------------------------------------------
// Row softmax, one wave32 per row (in place), float4 accesses.
// ---------------------------------------------------------------------------
__global__ void softmax_rows(float* __restrict__ S, int cols)
{
    const int wid  = (int)((blockIdx.x * blockDim.x + threadIdx.x) >> 5);
    const int lane = (int)(threadIdx.x & 31);
    float4* row = (float4*)(S + (long long)wid * cols);
    const int c4 = cols >> 2;

    float m = -3.402823466e+38f;
    for (int i = lane; i < c4; i += 32) {
        const float4 v = row[i];
        m = fmaxf(m, fmaxf(fmaxf(v.x, v.y), fmaxf(v.z, v.w)));
    }
#pragma unroll
    for (int s = 16; s > 0; s >>= 1) m = fmaxf(m, __shfl_xor(m, s, 32));

    float sum = 0.0f;
    for (int i = lane; i < c4; i += 32) {
        const float4 v = row[i];
        sum += __expf(v.x - m) + __expf(v.y - m) + __expf(v.z - m) + __expf(v.w - m);
    }
#pragma unroll
    for (int s = 16; s > 0; s >>= 1) sum += __shfl_xor(sum, s, 32);

    const float inv = 1.0f / sum;
    for (int i = lane; i < c4; i += 32) {
        float4 v = row[i];
        v.x = __expf(v.x - m) * inv; v.y = __expf(v.y - m) * inv;
        v.z = __expf(v.z - m) * inv; v.w = __expf(v.w - m) * inv;
        row[i] = v;
    }
}

// ---------------------------------------------------------------------------
// Host-side orchestration
// ---------------------------------------------------------------------------
extern "C" void kernel_launch(void* const* d_in, const int* in_sizes, int n_in,
                              void* d_out, int out_size, void* d_ws, size_t ws_size,
                              hipStream_t stream)
{
    (void)in_sizes; (void)n_in; (void)out_size; (void)ws_size;

    constexpr int B2   = 4;
    constexpr int CDIM = 512;
    constexpr int NPIX = 64 * 64;            // 4096
    constexpr int QK   = 64;

    const float* x  = (const float*)d_in[0];
    const float* Wq = (const float*)d_in[1];
    const float* bq = (const float*)d_in[2];
    const float* Wk = (const float*)d_in[3];
    const float* bk = (const float*)d_in[4];
    const float* Wv = (const float*)d_in[5];
    const float* bv = (const float*)d_in[6];
    float* out = (float*)d_out;

    const float* x_s = x;
    const float* x_t = x + (long long)B2 * CDIM * NPIX;

    float* ws  = (float*)d_ws;
    float* q_o = ws;                                     // [4, 64,4096]
    float* k_o = q_o + (long long)B2 * QK   * NPIX;      // [4, 64,4096]
    float* v_o = k_o + (long long)B2 * QK   * NPIX;      // [4,512,4096]
    float* S   = v_o + (long long)B2 * CDIM * NPIX;      // [4,4096,4096]

    const int THREADS = 256;
    auto blocks = [&](int O, int N) -> dim3 {
        return dim3((unsigned)(B2 * (O / TM) * (N / TN)));
    };

    // out[:4] = x_s
    (void)hipMemcpyAsync(out, x_s, (size_t)B2 * CDIM * NPIX * sizeof(float),
                         hipMemcpyDeviceToDevice, stream);

    // q[b,o,n] = Wq(o,c) x_s(b,c,n) + bq[o]      A: W[o][c] k-minor, B: x[c][n] n-minor
    wmma_gemm_tiled<<<blocks(QK, NPIX), THREADS, 0, stream>>>(
        Wq, CDIM, 0, 1,
        x_s, NPIX, (long long)CDIM * NPIX, 0,
        q_o, (long long)QK * NPIX,
        bq, nullptr, 0,
        QK, CDIM, NPIX, B2);

    // k[b,o,n] = Wk(o,c) x_t(b,c,n) + bk[o]
    wmma_gemm_tiled<<<blocks(QK, NPIX), THREADS, 0, stream>>>(
        Wk, CDIM, 0, 1,
        x_t, NPIX, (long long)CDIM * NPIX, 0,
        k_o, (long long)QK * NPIX,
        bk, nullptr, 0,
        QK, CDIM, NPIX, B2);

    // v[b,o,n] = Wv(o,c) x_t(b,c,n) + bv[o]
    wmma_gemm_tiled<<<blocks(CDIM, NPIX), THREADS, 0, stream>>>(
        Wv, CDIM, 0, 1,
        x_t, NPIX, (long long)CDIM * NPIX, 0,
        v_o, (long long)CDIM * NPIX,
        bv, nullptr, 0,
        CDIM, CDIM, NPIX, B2);

    // S[b,n,m] = sum_o q[b,o,n] k[b,o,m]         A: q[o][n] m-minor, B: k[o][m] n-minor
    wmma_gemm_tiled<<<blocks(NPIX, NPIX), THREADS, 0, stream>>>(
        q_o, NPIX, (long long)QK * NPIX, 0,
        k_o, NPIX, (long long)QK * NPIX, 0,
        S, (long long)NPIX * NPIX,
        nullptr, nullptr, 0,
        NPIX, QK, NPIX, B2);

    // softmax rows (b*n rows of length NPIX)
    {
        long long rows = (long long)B2 * NPIX;
        softmax_rows<<<dim3((unsigned)((rows * 32) / THREADS)), THREADS, 0, stream>>>(S, NPIX);
    }

    // out[b,c,n] = x_t + sum_m v[b,c,m] P[b,n,m] A: v[c][m] k-minor, B: S[n][m] k-minor
    wmma_gemm_tiled<<<blocks(CDIM, NPIX), THREADS, 0, stream>>>(
        v_o, NPIX, (long long)CDIM * NPIX, 1,
        S, NPIX, (long long)NPIX * NPIX, 1,
        out + (long long)B2 * CDIM * NPIX, (long long)CDIM * NPIX,
        nullptr,
        x_t, (long long)CDIM * NPIX,
        CDIM, NPIX, NPIX, B2);
}